// Net_1151051235746
// MI455X (gfx1250) — compile-verified
//
#include <hip/hip_runtime.h>
#include <hip/hip_bf16.h>

#define NN   50000
#define NE   300000
#define KIN  1433
#define HID  16
#define NOUT 7

typedef float v2f __attribute__((ext_vector_type(2)));
typedef float v8f __attribute__((ext_vector_type(8)));

// ---------------------------------------------------------------------------
// Kernel 1: P1 = features @ W1   (50000x1433 @ 1433x16), fp32 WMMA 16x16x4.
// One wave per 16-row tile; 625 blocks * 5 waves = 3125 tiles = 50000 rows.
// K loop unrolled 8x so each wave keeps ~2KB of loads in flight (MLP for the
// ~500-cycle HBM latency at ~3 resident waves/SIMD).
// ---------------------------------------------------------------------------
__global__ __launch_bounds__(160) void proj1_wmma(const float* __restrict__ A,
                                                  const float* __restrict__ W1,
                                                  float* __restrict__ P1) {
    const int lane = threadIdx.x & 31;
    const int wave = threadIdx.x >> 5;
    const int tile = blockIdx.x * 5 + wave;      // 0 .. 3124
    const int row0 = tile * 16;
    const int half = lane >> 4;                  // 0 -> K pair (0,1); 1 -> K pair (2,3)
    const int mn   = lane & 15;                  // M for A, N for B/C/D

    const float* __restrict__ arow = A  + (size_t)(row0 + mn) * KIN + half * 2;
    const float* __restrict__ bptr = W1 + (size_t)(half * 2) * HID + mn;

    v8f c = {0.f, 0.f, 0.f, 0.f, 0.f, 0.f, 0.f, 0.f};

    int k0 = 0;
    // 358 full K-steps cover k = 0 .. 1431; unroll 8 => 44 fat bodies + rem 6
    #pragma unroll 8
    for (; k0 < 1432; k0 += 4) {
        v2f a, b;
        a.x = arow[k0];
        a.y = arow[k0 + 1];
        b.x = bptr[(size_t)k0 * HID];
        b.y = bptr[(size_t)(k0 + 1) * HID];
        // D = A*B + C  (fp32, exact)
        c = __builtin_amdgcn_wmma_f32_16x16x4_f32(false, a, false, b,
                                                  (short)0, c, false, false);
    }
    // Tail: only k = 1432 is valid (KIN = 1433)
    {
        v2f a, b;
        a.x = (half == 0) ? arow[k0] : 0.f;                 // k = 1432 (half 0)
        a.y = 0.f;
        b.x = (half == 0) ? bptr[(size_t)k0 * HID] : 0.f;
        b.y = 0.f;
        c = __builtin_amdgcn_wmma_f32_16x16x4_f32(false, a, false, b,
                                                  (short)0, c, false, false);
    }

    // D layout: VGPR r -> row (r + 8*half), col = mn
    float* __restrict__ outp = P1 + (size_t)row0 * HID + mn;
    #pragma unroll
    for (int r = 0; r < 8; ++r) {
        outp[(size_t)(r + half * 8) * HID] = c[r];
    }
}

// ---------------------------------------------------------------------------
// Kernel 2: X[dst[e]] += P1[src[e]]  (16 lanes per edge, f32 atomics in L2)
// ---------------------------------------------------------------------------
__global__ __launch_bounds__(256) void scatter_h16(const float* __restrict__ P1,
                                                   const int* __restrict__ src,
                                                   const int* __restrict__ dst,
                                                   float* __restrict__ X) {
    int t = blockIdx.x * blockDim.x + threadIdx.x;
    if (t >= NE * HID) return;
    int e = t >> 4;
    int h = t & 15;
    int s = src[e];
    int d = dst[e];
    atomicAdd(&X[(size_t)d * HID + h], P1[(size_t)s * HID + h]);
}

// ---------------------------------------------------------------------------
// Kernel 3: x = relu(X + b1); P2 = x @ W2; out = P2 + b2 (init for scatter)
// One thread per node (16x7 GEMM is tiny; fuse everything).
// ---------------------------------------------------------------------------
__global__ __launch_bounds__(256) void proj2_fused(const float* __restrict__ X,
                                                   const float* __restrict__ b1,
                                                   const float* __restrict__ W2,
                                                   const float* __restrict__ b2,
                                                   float* __restrict__ P2,
                                                   float* __restrict__ out) {
    int v = blockIdx.x * blockDim.x + threadIdx.x;
    if (v >= NN) return;
    float xr[HID];
    const float4* __restrict__ xp = (const float4*)(X + (size_t)v * HID);
    #pragma unroll
    for (int q = 0; q < 4; ++q) {
        float4 t = xp[q];
        xr[q * 4 + 0] = fmaxf(t.x + b1[q * 4 + 0], 0.f);
        xr[q * 4 + 1] = fmaxf(t.y + b1[q * 4 + 1], 0.f);
        xr[q * 4 + 2] = fmaxf(t.z + b1[q * 4 + 2], 0.f);
        xr[q * 4 + 3] = fmaxf(t.w + b1[q * 4 + 3], 0.f);
    }
    #pragma unroll
    for (int o = 0; o < NOUT; ++o) {
        float acc = 0.f;
        #pragma unroll
        for (int h = 0; h < HID; ++h)
            acc = fmaf(xr[h], W2[h * NOUT + o], acc);
        P2[(size_t)v * NOUT + o]  = acc;
        out[(size_t)v * NOUT + o] = acc + b2[o];
    }
}

// ---------------------------------------------------------------------------
// Kernel 4: out[dst[e]] += P2[src[e]]  (8 lanes per edge, 7 active)
// ---------------------------------------------------------------------------
__global__ __launch_bounds__(256) void scatter_h7(const float* __restrict__ P2,
                                                  const int* __restrict__ src,
                                                  const int* __restrict__ dst,
                                                  float* __restrict__ out) {
    int t = blockIdx.x * blockDim.x + threadIdx.x;
    if (t >= NE * 8) return;
    int e = t >> 3;
    int h = t & 7;
    if (h < NOUT) {
        atomicAdd(&out[(size_t)dst[e] * NOUT + h], P2[(size_t)src[e] * NOUT + h]);
    }
}

// ---------------------------------------------------------------------------
extern "C" void kernel_launch(void* const* d_in, const int* in_sizes, int n_in,
                              void* d_out, int out_size, void* d_ws, size_t ws_size,
                              hipStream_t stream) {
    (void)in_sizes; (void)n_in; (void)out_size; (void)ws_size;

    const float* features = (const float*)d_in[0];
    const int*   src      = (const int*)  d_in[1];
    const int*   dst      = (const int*)  d_in[2];
    const float* W1       = (const float*)d_in[3];
    const float* b1       = (const float*)d_in[4];
    const float* W2       = (const float*)d_in[5];
    const float* b2       = (const float*)d_in[6];
    float*       out      = (float*)d_out;

    // Workspace layout (f32): P1 [NN*HID] | X [NN*HID] | P2 [NN*NOUT]  ~7.8 MB
    float* P1 = (float*)d_ws;
    float* X  = P1 + (size_t)NN * HID;
    float* P2 = X  + (size_t)NN * HID;

    // 1) P1 = features @ W1  (WMMA fp32)
    proj1_wmma<<<625, 160, 0, stream>>>(features, W1, P1);

    // 2) X = P1 (self term), then X[dst] += P1[src]
    hipMemcpyAsync(X, P1, (size_t)NN * HID * sizeof(float),
                   hipMemcpyDeviceToDevice, stream);
    scatter_h16<<<(NE * HID) / 256, 256, 0, stream>>>(P1, src, dst, X);

    // 3) x = relu(X+b1); P2 = x@W2; out = P2 + b2
    proj2_fused<<<(NN + 255) / 256, 256, 0, stream>>>(X, b1, W2, b2, P2, out);

    // 4) out[dst] += P2[src]
    scatter_h7<<<(NE * 8) / 256, 256, 0, stream>>>(P2, src, dst, out);
}